// THPN_31379031065271
// MI455X (gfx1250) — compile-verified
//
#include <hip/hip_runtime.h>
#include <math.h>

#define V_   32000
#define E_   512
#define B_   64
#define M_   256
#define S_   6
#define R_   10
#define NH_  3

typedef __attribute__((ext_vector_type(2))) float v2f;
typedef __attribute__((ext_vector_type(4))) float v4f;
typedef __attribute__((ext_vector_type(8))) float v8f;

// Two chained 16x16x4 f32 WMMAs covering 8 K-values held in one b128/lane.
// K-set of wmma #0 = {k0,k0+1,k0+4,k0+5}, #1 = {k0+2,k0+3,k0+6,k0+7};
// A and B fragments use identical lane->K assignment and WMMA accumulation is
// K-order independent, so any consistent partition is valid.
__device__ __forceinline__ v8f wmma_pair(v4f a, v4f b, v8f c)
{
  v2f al; al.x = a.x; al.y = a.y;
  v2f ah; ah.x = a.z; ah.y = a.w;
  v2f bl; bl.x = b.x; bl.y = b.y;
  v2f bh; bh.x = b.z; bh.y = b.w;
  c = __builtin_amdgcn_wmma_f32_16x16x4_f32(false, al, false, bl, (short)0, c, false, false);
  c = __builtin_amdgcn_wmma_f32_16x16x4_f32(false, ah, false, bh, (short)0, c, false, false);
  return c;
}

// ---------------------------------------------------------------------------
// Generic f32 WMMA GEMM:  Y = act(X * W^T + bias)
//   X: (Nr, K) row-major,  W: (No, K) row-major,  Y: (Nr, No)
//   grid = (No/32, Nr/64), block = 32 threads (one wave).
//   64x32 output tile per wave (8x v8f accumulators): per 8-K group we issue
//   6 b128 loads and 16 WMMAs (8 independent accumulation chains), with
//   manual double buffering so HBM reads overlap the matrix pipe.
//   Requires K % 8 == 0 (all K here are 512/1024/1536).
// ---------------------------------------------------------------------------
__global__ void wmma_gemm_f32(const float* __restrict__ X,
                              const float* __restrict__ W,
                              const float* __restrict__ bias,
                              float* __restrict__ Y,
                              int Nr, int No, int K, int act)
{
  const int col0 = blockIdx.x * 32;
  const int row0 = blockIdx.y * 64;
  const int lane = threadIdx.x;
  const int nn = lane & 15;           // col (B/C/D) or row (A) within tile
  const int kq = (lane >> 4) * 4;     // K sub-offset: 0 (lanes 0-15) or 4

  v8f c00 = {}; v8f c01 = {}; v8f c02 = {}; v8f c03 = {};   // col tile 0
  v8f c10 = {}; v8f c11 = {}; v8f c12 = {}; v8f c13 = {};   // col tile 1

  const float* Wp0 = W + (size_t)(col0 + nn) * K + kq;
  const float* Wp1 = W + (size_t)(col0 + 16 + nn) * K + kq;
  const float* Xp  = X + (size_t)(row0 + nn) * K + kq;
  const size_t rstep = (size_t)16 * K;

  // prologue loads (k0 = 0)
  v4f b0 = *(const v4f*)(Wp0);
  v4f b1 = *(const v4f*)(Wp1);
  v4f a0 = *(const v4f*)(Xp);
  v4f a1 = *(const v4f*)(Xp + rstep);
  v4f a2 = *(const v4f*)(Xp + 2 * rstep);
  v4f a3 = *(const v4f*)(Xp + 3 * rstep);

  int k0 = 0;
  for (; k0 + 8 < K; k0 += 8) {
    // prefetch next 8-K group while current WMMAs run
    v4f b0n = *(const v4f*)(Wp0 + k0 + 8);
    v4f b1n = *(const v4f*)(Wp1 + k0 + 8);
    v4f a0n = *(const v4f*)(Xp + k0 + 8);
    v4f a1n = *(const v4f*)(Xp + k0 + 8 + rstep);
    v4f a2n = *(const v4f*)(Xp + k0 + 8 + 2 * rstep);
    v4f a3n = *(const v4f*)(Xp + k0 + 8 + 3 * rstep);

    c00 = wmma_pair(a0, b0, c00);
    c01 = wmma_pair(a1, b0, c01);
    c02 = wmma_pair(a2, b0, c02);
    c03 = wmma_pair(a3, b0, c03);
    c10 = wmma_pair(a0, b1, c10);
    c11 = wmma_pair(a1, b1, c11);
    c12 = wmma_pair(a2, b1, c12);
    c13 = wmma_pair(a3, b1, c13);

    b0 = b0n; b1 = b1n; a0 = a0n; a1 = a1n; a2 = a2n; a3 = a3n;
  }
  // epilogue (last 8-K group)
  c00 = wmma_pair(a0, b0, c00);
  c01 = wmma_pair(a1, b0, c01);
  c02 = wmma_pair(a2, b0, c02);
  c03 = wmma_pair(a3, b0, c03);
  c10 = wmma_pair(a0, b1, c10);
  c11 = wmma_pair(a1, b1, c11);
  c12 = wmma_pair(a2, b1, c12);
  c13 = wmma_pair(a3, b1, c13);

  const float bv0  = bias[col0 + nn];
  const float bv1  = bias[col0 + 16 + nn];
  const int  rbase = row0 + ((lane >> 4) * 8);
  #pragma unroll
  for (int j = 0; j < 8; ++j) {
    float y00 = c00[j] + bv0, y01 = c01[j] + bv0, y02 = c02[j] + bv0, y03 = c03[j] + bv0;
    float y10 = c10[j] + bv1, y11 = c11[j] + bv1, y12 = c12[j] + bv1, y13 = c13[j] + bv1;
    if (act) {
      y00 = tanhf(y00); y01 = tanhf(y01); y02 = tanhf(y02); y03 = tanhf(y03);
      y10 = tanhf(y10); y11 = tanhf(y11); y12 = tanhf(y12); y13 = tanhf(y13);
    }
    Y[(size_t)(rbase +      j) * No + col0 + nn]      = y00;
    Y[(size_t)(rbase + 16 + j) * No + col0 + nn]      = y01;
    Y[(size_t)(rbase + 32 + j) * No + col0 + nn]      = y02;
    Y[(size_t)(rbase + 48 + j) * No + col0 + nn]      = y03;
    Y[(size_t)(rbase +      j) * No + col0 + 16 + nn] = y10;
    Y[(size_t)(rbase + 16 + j) * No + col0 + 16 + nn] = y11;
    Y[(size_t)(rbase + 32 + j) * No + col0 + 16 + nn] = y12;
    Y[(size_t)(rbase + 48 + j) * No + col0 + 16 + nn] = y13;
  }
}

// ---------------------------------------------------------------------------
// x = C0[enc_query]   (B, E)
// ---------------------------------------------------------------------------
__global__ void gather_xq(const int* __restrict__ q, const float* __restrict__ C0,
                          float* __restrict__ xq)
{
  int b = blockIdx.x;
  int idx = q[b];
  for (int e = threadIdx.x; e < E_; e += blockDim.x)
    xq[b * E_ + e] = C0[(size_t)idx * E_ + e];
}

// ---------------------------------------------------------------------------
// GRU elementwise gate math; also seeds u_story / u_ans with hnew.
// ---------------------------------------------------------------------------
__global__ void gru_elem(const float* __restrict__ gi, const float* __restrict__ gh,
                         const float* __restrict__ h0,
                         float* __restrict__ hnew, float* __restrict__ u1,
                         float* __restrict__ u2)
{
  int i = blockIdx.x * blockDim.x + threadIdx.x;
  if (i >= B_ * E_) return;
  int b = i / E_, e = i % E_;
  const float* gib = gi + (size_t)b * 3 * E_;
  const float* ghb = gh + (size_t)b * 3 * E_;
  float r = 1.f / (1.f + expf(-(gib[e]          + ghb[e])));
  float z = 1.f / (1.f + expf(-(gib[E_ + e]     + ghb[E_ + e])));
  float n = tanhf(gib[2 * E_ + e] + r * ghb[2 * E_ + e]);
  float h = (1.f - z) * n + z * h0[i];
  hnew[i] = h; u1[i] = h; u2[i] = h;
}

// ---------------------------------------------------------------------------
// rows r*B+b of [ CA0[ans[b][r]] , h0[b] ]  -> (R*B, 2E)
// ---------------------------------------------------------------------------
__global__ void build_gate_cat(const int* __restrict__ ans, const float* __restrict__ CA0,
                               const float* __restrict__ h0, float* __restrict__ out)
{
  int row = blockIdx.x;            // r*B + b
  int r = row / B_, b = row % B_;
  int idx = ans[b * R_ + r];
  const float* arow = CA0 + (size_t)idx * E_;
  float* o = out + (size_t)row * (2 * E_);
  for (int e = threadIdx.x; e < E_; e += blockDim.x) {
    o[e]      = arow[e];
    o[E_ + e] = h0[b * E_ + e];
  }
}

// ---------------------------------------------------------------------------
// gate_hidden[b][e] = (sum_r softmax_e(logits[r*B+b]))[e] * h0[b][e]
// ---------------------------------------------------------------------------
__global__ void gate_reduce(const float* __restrict__ logits, const float* __restrict__ h0,
                            float* __restrict__ gate)
{
  __shared__ float red[256];
  int b = blockIdx.x, t = threadIdx.x;
  float acc0 = 0.f, acc1 = 0.f;
  for (int r = 0; r < R_; ++r) {
    const float* row = logits + (size_t)(r * B_ + b) * E_;
    float l0 = row[t], l1 = row[t + 256];
    red[t] = fmaxf(l0, l1); __syncthreads();
    for (int s = 128; s > 0; s >>= 1) { if (t < s) red[t] = fmaxf(red[t], red[t + s]); __syncthreads(); }
    float mx = red[0]; __syncthreads();
    float e0 = expf(l0 - mx), e1 = expf(l1 - mx);
    red[t] = e0 + e1; __syncthreads();
    for (int s = 128; s > 0; s >>= 1) { if (t < s) red[t] += red[t + s]; __syncthreads(); }
    float inv = 1.f / red[0]; __syncthreads();
    acc0 += e0 * inv; acc1 += e1 * inv;
  }
  gate[b * E_ + t]       = acc0 * h0[b * E_ + t];
  gate[b * E_ + t + 256] = acc1 * h0[b * E_ + t + 256];
}

// ---------------------------------------------------------------------------
__global__ void concat2(const float* __restrict__ A, const float* __restrict__ Bs,
                        float* __restrict__ out, int E1, int E2)
{
  int row = blockIdx.x;
  float* o = out + (size_t)row * (E1 + E2);
  for (int e = threadIdx.x; e < E1; e += blockDim.x) o[e]      = A[(size_t)row * E1 + e];
  for (int e = threadIdx.x; e < E2; e += blockDim.x) o[E1 + e] = Bs[(size_t)row * E2 + e];
}

__global__ void concat3(const float* __restrict__ A, const float* __restrict__ Bs,
                        const float* __restrict__ Cs, float* __restrict__ out)
{
  int row = blockIdx.x;
  float* o = out + (size_t)row * (3 * E_);
  for (int e = threadIdx.x; e < E_; e += blockDim.x) {
    o[e]           = A[(size_t)row * E_ + e];
    o[E_ + e]      = Bs[(size_t)row * E_ + e];
    o[2 * E_ + e]  = Cs[(size_t)row * E_ + e];
  }
}

// rows b*NH+n of [ hnew[b], hop_inf[n][b] ] -> (B*NH, 2E)
__global__ void hopcat(const float* __restrict__ hnew, const float* __restrict__ hop_inf,
                       float* __restrict__ out)
{
  int row = blockIdx.x;           // b*NH + n
  int b = row / NH_, n = row % NH_;
  float* o = out + (size_t)row * (2 * E_);
  for (int e = threadIdx.x; e < E_; e += blockDim.x) {
    o[e]      = hnew[b * E_ + e];
    o[E_ + e] = hop_inf[(size_t)n * B_ * E_ + b * E_ + e];
  }
}

// escore = v . energy ; softmax over NH ; context = sum_n a_n * hop_inf[n][b]
__global__ void hop_attn(const float* __restrict__ energy, const float* __restrict__ v,
                         const float* __restrict__ hop_inf, float* __restrict__ context)
{
  __shared__ float red[256];
  __shared__ float a[NH_];
  int b = blockIdx.x, t = threadIdx.x;
  float esc[NH_];
  for (int n = 0; n < NH_; ++n) {
    const float* row = energy + (size_t)(b * NH_ + n) * E_;
    red[t] = row[t] * v[t] + row[t + 256] * v[t + 256];
    __syncthreads();
    for (int s = 128; s > 0; s >>= 1) { if (t < s) red[t] += red[t + s]; __syncthreads(); }
    esc[n] = red[0]; __syncthreads();
  }
  if (t == 0) {
    float mx = fmaxf(esc[0], fmaxf(esc[1], esc[2]));
    float s = 0.f;
    for (int n = 0; n < NH_; ++n) { a[n] = expf(esc[n] - mx); s += a[n]; }
    for (int n = 0; n < NH_; ++n) a[n] /= s;
  }
  __syncthreads();
  for (int k = 0; k < 2; ++k) {
    int e = t + k * 256;
    float c = 0.f;
    for (int n = 0; n < NH_; ++n) c += a[n] * hop_inf[(size_t)n * B_ * E_ + b * E_ + e];
    context[b * E_ + e] = c;
  }
}

// ---------------------------------------------------------------------------
// prob_lg[b][m] = sum_s dot(Ch[story[b][m][s]], u[b])
// one block of 128 threads per (b,m); b128 loads (4 floats/lane).
// ---------------------------------------------------------------------------
__global__ void story_problg(const int* __restrict__ story, const float* __restrict__ Ch,
                             const float* __restrict__ u, float* __restrict__ out)
{
  __shared__ float red[128];
  int bm = blockIdx.x;            // b*M + m
  int b  = bm / M_;
  int t  = threadIdx.x;
  const int* srow = story + (size_t)bm * S_;
  const v4f* ub4  = (const v4f*)(u + (size_t)b * E_);
  v4f uv = ub4[t];
  float p = 0.f;
  for (int s = 0; s < S_; ++s) {
    const v4f* crow = (const v4f*)(Ch + (size_t)srow[s] * E_);
    v4f cv = crow[t];
    p += cv.x * uv.x + cv.y * uv.y + cv.z * uv.z + cv.w * uv.w;
  }
  red[t] = p; __syncthreads();
  for (int s2 = 64; s2 > 0; s2 >>= 1) { if (t < s2) red[t] += red[t + s2]; __syncthreads(); }
  if (t == 0) out[bm] = red[0];
}

// softmax over M=256 per row b
__global__ void softmax256(const float* __restrict__ logits, float* __restrict__ prob)
{
  __shared__ float red[256];
  int b = blockIdx.x, t = threadIdx.x;
  float l = logits[b * M_ + t];
  red[t] = l; __syncthreads();
  for (int s = 128; s > 0; s >>= 1) { if (t < s) red[t] = fmaxf(red[t], red[t + s]); __syncthreads(); }
  float mx = red[0]; __syncthreads();
  float e = expf(l - mx);
  red[t] = e; __syncthreads();
  for (int s = 128; s > 0; s >>= 1) { if (t < s) red[t] += red[t + s]; __syncthreads(); }
  prob[b * M_ + t] = e / red[0];
}

// o_k[b][e] = sum_m prob[b][m] * sum_s Cn[story[b][m][s]][e];  u += o_k
// grid (B, 4), one wave per block; each lane accumulates a float4 (e-chunk 128)
__global__ void story_ok_update(const int* __restrict__ story, const float* __restrict__ Cn,
                                const float* __restrict__ prob,
                                float* __restrict__ o_k, float* __restrict__ u)
{
  __shared__ float ps[M_];
  int b = blockIdx.x;
  int t = threadIdx.x;                 // 0..31
  for (int i = t; i < M_; i += 32) ps[i] = prob[b * M_ + i];
  __syncthreads();
  int e4 = blockIdx.y * 32 + t;        // float4 index within the E=512 row
  v4f acc = {0.f, 0.f, 0.f, 0.f};
  const int* srow = story + (size_t)b * M_ * S_;
  for (int m = 0; m < M_; ++m) {
    float pm = ps[m];
    const int* sm = srow + m * S_;
    #pragma unroll
    for (int s = 0; s < S_; ++s) {
      const v4f* row = (const v4f*)(Cn + (size_t)sm[s] * E_);
      v4f cv = row[e4];
      acc += cv * pm;
    }
  }
  v4f* ok4 = (v4f*)(o_k + (size_t)b * E_);
  v4f* u4  = (v4f*)(u   + (size_t)b * E_);
  ok4[e4] = acc;
  u4[e4]  = u4[e4] + acc;
}

// ---------------------------------------------------------------------------
// Fused answer-memory hop: logits -> (softmax -> o -> u update if do_update)
// ---------------------------------------------------------------------------
__global__ void ans_hop(const int* __restrict__ ans, const float* __restrict__ CAh,
                        const float* __restrict__ CAn,
                        float* __restrict__ u, float* __restrict__ out_logits, int do_update)
{
  __shared__ float red[256];
  __shared__ float pr[R_];
  __shared__ int   idxs[R_];
  int b = blockIdx.x, t = threadIdx.x;
  float* ub = u + (size_t)b * E_;
  if (t < R_) idxs[t] = ans[b * R_ + t];
  __syncthreads();
  float lg[R_];
  for (int r = 0; r < R_; ++r) {
    const float* row = CAh + (size_t)idxs[r] * E_;
    red[t] = row[t] * ub[t] + row[t + 256] * ub[t + 256];
    __syncthreads();
    for (int s = 128; s > 0; s >>= 1) { if (t < s) red[t] += red[t + s]; __syncthreads(); }
    lg[r] = red[0]; __syncthreads();
    if (t == 0) out_logits[b * R_ + r] = lg[r];
  }
  if (!do_update) return;
  if (t == 0) {
    float mx = lg[0];
    for (int r = 1; r < R_; ++r) mx = fmaxf(mx, lg[r]);
    float s = 0.f;
    for (int r = 0; r < R_; ++r) { pr[r] = expf(lg[r] - mx); s += pr[r]; }
    for (int r = 0; r < R_; ++r) pr[r] /= s;
  }
  __syncthreads();
  for (int k = 0; k < 2; ++k) {
    int e = t + k * 256;
    float o = 0.f;
    for (int r = 0; r < R_; ++r) o += pr[r] * CAn[(size_t)idxs[r] * E_ + e];
    ub[e] += o;
  }
}

// ---------------------------------------------------------------------------
extern "C" void kernel_launch(void* const* d_in, const int* in_sizes, int n_in,
                              void* d_out, int out_size, void* d_ws, size_t ws_size,
                              hipStream_t stream)
{
  (void)in_sizes; (void)n_in; (void)out_size; (void)ws_size;

  const int*   story     = (const int*)  d_in[0];
  const int*   ans_story = (const int*)  d_in[1];
  const int*   enc_query = (const int*)  d_in[2];
  const float* h0        = (const float*)d_in[3];   // last_hidden[0], (B,E)
  const float* hop_inf   = (const float*)d_in[4];
  const float* C         = (const float*)d_in[5];
  const float* CA        = (const float*)d_in[6];
  const float* W2_w  = (const float*)d_in[7];
  const float* W2_b  = (const float*)d_in[8];
  const float* W3_w  = (const float*)d_in[9];
  const float* W3_b  = (const float*)d_in[10];
  const float* W1_w  = (const float*)d_in[11];
  const float* W1_b  = (const float*)d_in[12];
  const float* Wg_w  = (const float*)d_in[13];
  const float* Wg_b  = (const float*)d_in[14];
  const float* Wgg_w = (const float*)d_in[15];
  const float* Wgg_b = (const float*)d_in[16];
  const float* concat_w = (const float*)d_in[17];
  const float* concat_b = (const float*)d_in[18];
  const float* out_w = (const float*)d_in[19];
  const float* out_b = (const float*)d_in[20];
  const float* vvec  = (const float*)d_in[21];
  const float* gru_Wih = (const float*)d_in[22];
  const float* gru_Whh = (const float*)d_in[23];
  const float* gru_bih = (const float*)d_in[24];
  const float* gru_bhh = (const float*)d_in[25];

  const size_t VE = (size_t)V_ * E_;

  // output regions (return order: p_ptr, p_vocab, p_ans, hidden_res)
  float* out       = (float*)d_out;
  float* p_ptr     = out;
  float* p_vocab   = out + (size_t)B_ * M_;
  float* p_ans     = p_vocab + (size_t)B_ * V_;
  float* hid_res   = p_ans + (size_t)B_ * R_;

  // workspace carve (floats), all chunks 16B-aligned
  float* ws = (float*)d_ws;
  size_t off = 0;
  float* xq        = ws + off; off += (size_t)B_ * E_;
  float* gi        = ws + off; off += (size_t)B_ * 3 * E_;
  float* gh        = ws + off; off += (size_t)B_ * 3 * E_;
  float* hnew      = ws + off; off += (size_t)B_ * E_;
  float* gate_cat  = ws + off; off += (size_t)R_ * B_ * 2 * E_;
  float* sm_logits = ws + off; off += (size_t)R_ * B_ * E_;
  float* gate      = ws + off; off += (size_t)B_ * E_;
  float* catA      = ws + off; off += (size_t)B_ * 2 * E_;
  float* tmpg      = ws + off; off += (size_t)B_ * E_;
  float* hidden_g  = ws + off; off += (size_t)B_ * E_;
  float* catH      = ws + off; off += (size_t)B_ * NH_ * 2 * E_;
  float* energy    = ws + off; off += (size_t)B_ * NH_ * E_;
  float* context   = ws + off; off += (size_t)B_ * E_;
  float* cat3b     = ws + off; off += (size_t)B_ * 3 * E_;
  float* conc_out  = ws + off; off += (size_t)B_ * E_;
  float* u_story   = ws + off; off += (size_t)B_ * E_;
  float* u_ans     = ws + off; off += (size_t)B_ * E_;
  float* problg    = ws + off; off += (size_t)B_ * M_;
  float* prob      = ws + off; off += (size_t)B_ * M_;
  float* o_k       = ws + off; off += (size_t)B_ * E_;
  float* cat_w1    = ws + off; off += (size_t)B_ * 2 * E_;
  float* ans_lg    = ws + off; off += (size_t)B_ * R_;

  // ---- GRU: x = C0[enc_query]; gi = x@Wih^T+bih ; gh = h0@Whh^T+bhh ----
  gather_xq<<<B_, 256, 0, stream>>>(enc_query, C, xq);
  wmma_gemm_f32<<<dim3(3 * E_ / 32, 1), 32, 0, stream>>>(xq, gru_Wih, gru_bih, gi, B_, 3 * E_, E_, 0);
  wmma_gemm_f32<<<dim3(3 * E_ / 32, 1), 32, 0, stream>>>(h0, gru_Whh, gru_bhh, gh, B_, 3 * E_, E_, 0);
  gru_elem<<<(B_ * E_) / 256, 256, 0, stream>>>(gi, gh, h0, hnew, u_story, u_ans);

  // ---- gate_hidden (answer attention over h0) ----
  build_gate_cat<<<R_ * B_, 256, 0, stream>>>(ans_story, CA, h0, gate_cat);
  wmma_gemm_f32<<<dim3(E_ / 32, (R_ * B_) / 64), 32, 0, stream>>>(gate_cat, W2_w, W2_b, sm_logits,
                                                                  R_ * B_, E_, 2 * E_, 0);
  gate_reduce<<<B_, 256, 0, stream>>>(sm_logits, h0, gate);

  // ---- hidden_g = tanh([hnew,gate]@Wg^T+b) @ Wgg^T + b ----
  concat2<<<B_, 256, 0, stream>>>(hnew, gate, catA, E_, E_);
  wmma_gemm_f32<<<dim3(E_ / 32, 1), 32, 0, stream>>>(catA, Wg_w, Wg_b, tmpg, B_, E_, 2 * E_, 1);
  wmma_gemm_f32<<<dim3(E_ / 32, 1), 32, 0, stream>>>(tmpg, Wgg_w, Wgg_b, hidden_g, B_, E_, E_, 0);

  // ---- hop attention ----
  hopcat<<<B_ * NH_, 256, 0, stream>>>(hnew, hop_inf, catH);
  wmma_gemm_f32<<<dim3(E_ / 32, (B_ * NH_) / 64), 32, 0, stream>>>(catH, W3_w, W3_b, energy,
                                                                   B_ * NH_, E_, 2 * E_, 1);
  hop_attn<<<B_, 256, 0, stream>>>(energy, vvec, hop_inf, context);

  // ---- hidden_res ----
  concat3<<<B_, 256, 0, stream>>>(hnew, hidden_g, context, cat3b);
  wmma_gemm_f32<<<dim3(E_ / 32, 1), 32, 0, stream>>>(cat3b, concat_w, concat_b, conc_out, B_, E_, 3 * E_, 1);
  wmma_gemm_f32<<<dim3(E_ / 32, 1), 32, 0, stream>>>(conc_out, out_w, out_b, hid_res, B_, E_, E_, 0);

  // ---- story memory hops ----
  // hop 0
  story_problg<<<B_ * M_, 128, 0, stream>>>(story, C + 0 * VE, u_story, problg);
  softmax256<<<B_, 256, 0, stream>>>(problg, prob);
  story_ok_update<<<dim3(B_, 4), 32, 0, stream>>>(story, C + 1 * VE, prob, o_k, u_story);
  // p_vocab = [hnew, o_k] @ W1^T + b   (64 x 32000 x 1024, single pass over W1)
  concat2<<<B_, 256, 0, stream>>>(hnew, o_k, cat_w1, E_, E_);
  wmma_gemm_f32<<<dim3(V_ / 32, 1), 32, 0, stream>>>(cat_w1, W1_w, W1_b, p_vocab, B_, V_, 2 * E_, 0);
  // hop 1
  story_problg<<<B_ * M_, 128, 0, stream>>>(story, C + 1 * VE, u_story, problg);
  softmax256<<<B_, 256, 0, stream>>>(problg, prob);
  story_ok_update<<<dim3(B_, 4), 32, 0, stream>>>(story, C + 2 * VE, prob, o_k, u_story);
  // hop 2 (only logits needed -> p_ptr)
  story_problg<<<B_ * M_, 128, 0, stream>>>(story, C + 2 * VE, u_story, p_ptr);

  // ---- answer memory hops (fused) ----
  ans_hop<<<B_, 256, 0, stream>>>(ans_story, CA + 0 * VE, CA + 1 * VE, u_ans, ans_lg, 1);
  ans_hop<<<B_, 256, 0, stream>>>(ans_story, CA + 1 * VE, CA + 2 * VE, u_ans, ans_lg, 1);
  ans_hop<<<B_, 256, 0, stream>>>(ans_story, CA + 2 * VE, CA + 2 * VE, u_ans, p_ans, 0);
}